// GraphSAGE_38543036514868
// MI455X (gfx1250) — compile-verified
//
#include <hip/hip_runtime.h>
#include <hip/hip_bf16.h>

typedef __attribute__((ext_vector_type(2))) float v2f;
typedef __attribute__((ext_vector_type(8))) float v8f;

#define N_NODES 100000
#define N_EDGES 800000
#define D_FEAT  128   // D_IN == D_HID == 128
#define LDS_STR 132   // padded row stride (floats): 4-float skew -> conflict-free b64 LDS reads

// ---------------------------------------------------------------------------
// Zero a buffer (float4 granularity, grid-stride)
// ---------------------------------------------------------------------------
__global__ void zero4_kernel(float4* __restrict__ p, int n4) {
    int i = blockIdx.x * blockDim.x + threadIdx.x;
    for (; i < n4; i += gridDim.x * blockDim.x) {
        p[i] = make_float4(0.f, 0.f, 0.f, 0.f);
    }
}

// ---------------------------------------------------------------------------
// Edge scatter: one wave32 per edge. Each lane gathers float4 of the source
// row and atomically accumulates into the destination row (L2 atomic units).
// ---------------------------------------------------------------------------
template<bool WITH_DEG>
__global__ void aggregate_kernel(const float* __restrict__ X,
                                 const int*   __restrict__ src,
                                 const int*   __restrict__ dst,
                                 float*       __restrict__ ns,
                                 float*       __restrict__ deg,
                                 int nEdges) {
    long long t = (long long)blockIdx.x * blockDim.x + threadIdx.x;
    int e    = (int)(t >> 5);
    int lane = (int)(t & 31);
    if (e >= nEdges) return;
    int s = src[e];
    int d = dst[e];
    const float4 v = *(const float4*)&X[(long long)s * D_FEAT + lane * 4];
    float* p = &ns[(long long)d * D_FEAT + lane * 4];
    atomicAdd(p + 0, v.x);
    atomicAdd(p + 1, v.y);
    atomicAdd(p + 2, v.z);
    atomicAdd(p + 3, v.w);
    if (WITH_DEG && lane == 0) atomicAdd(&deg[d], 1.0f);
}

// ---------------------------------------------------------------------------
// SAGE layer: out[m,n] = act( X[m,:] @ Wself[:,n] + (ns[m,:]/max(deg,1)) @ Wneigh[:,n] )
//
// Block = MT*16 nodes, DOUT/16 waves. Register-level M-blocking: each wave
// keeps one B (weight) fragment pair in VGPRs and reuses it across MT A-tiles
// -> per K-step: 4 global B loads + 2*MT LDS A loads feed 2*MT WMMAs, with MT
// independent accumulator chains for WMMA-pipe ILP.
// fp32 WMMA (V_WMMA_F32_16X16X4_F32): fp32 in / fp32 accumulate = reference
// precision; at ~10 GFLOP total the GEMMs are nowhere near the matrix-op
// ceiling, so there is no reason to drop to bf16/f16.
// ---------------------------------------------------------------------------
template<int DOUT, int MT, bool RELU>
__global__ void sage_layer_kernel(const float* __restrict__ X,
                                  const float* __restrict__ ns,
                                  const float* __restrict__ deg,
                                  const float* __restrict__ Wself,
                                  const float* __restrict__ Wneigh,
                                  float*       __restrict__ out) {
    constexpr int MROWS = MT * 16;
    __shared__ float sSelf[MROWS * LDS_STR];
    __shared__ float sNeigh[MROWS * LDS_STR];

    const int base = blockIdx.x * MROWS;

    // Stage MROWSx128 self tile and degree-normalized neigh tile into LDS.
    // Coalesced global_load_b128 / ds_store_b128.
    for (int i = threadIdx.x; i < MROWS * 32; i += blockDim.x) {
        int row = i >> 5;             // 0..MROWS-1
        int c4  = (i & 31) * 4;       // 0..124
        long long g = (long long)(base + row) * D_FEAT + c4;
        float4 fs = *(const float4*)&X[g];
        float4 fn = *(const float4*)&ns[g];
        float inv = 1.0f / fmaxf(deg[base + row], 1.0f);
        fn.x *= inv; fn.y *= inv; fn.z *= inv; fn.w *= inv;
        *(float4*)&sSelf[row * LDS_STR + c4]  = fs;
        *(float4*)&sNeigh[row * LDS_STR + c4] = fn;
    }
    __syncthreads();

    const int lane = threadIdx.x & 31;
    const int wid  = threadIdx.x >> 5;
    const int lo   = lane & 15;          // A: M-row index; B/D: tile-local N col
    const int hi   = lane >> 4;          // selects K pair (A/B) and M half (D)
    const int n    = wid * 16 + lo;      // global output column

    v8f acc[MT] = {};
#pragma unroll 4
    for (int kk = 0; kk < D_FEAT; kk += 4) {
        const int kb = kk + 2 * hi;
        // B fragments (4x16 f32): lane holds W[kb][n], W[kb+1][n]; reused MT times
        v2f bS, bN;
        bS[0] = Wself [(kb    ) * DOUT + n];
        bS[1] = Wself [(kb + 1) * DOUT + n];
        bN[0] = Wneigh[(kb    ) * DOUT + n];
        bN[1] = Wneigh[(kb + 1) * DOUT + n];
#pragma unroll
        for (int t = 0; t < MT; ++t) {
            // A fragments (16x4 f32): lane holds A[lo][kb], A[lo][kb+1] -> ds_load_b64
            v2f aS = *(const v2f*)&sSelf [(t * 16 + lo) * LDS_STR + kb];
            v2f aN = *(const v2f*)&sNeigh[(t * 16 + lo) * LDS_STR + kb];
            acc[t] = __builtin_amdgcn_wmma_f32_16x16x4_f32(false, aS, false, bS,
                                                           (short)0, acc[t], false, false);
            acc[t] = __builtin_amdgcn_wmma_f32_16x16x4_f32(false, aN, false, bN,
                                                           (short)0, acc[t], false, false);
        }
    }

    // D layout: VGPR r -> M = t*16 + r + 8*hi, N = lo
#pragma unroll
    for (int t = 0; t < MT; ++t) {
#pragma unroll
        for (int r = 0; r < 8; ++r) {
            int m = base + t * 16 + r + 8 * hi;
            float v = acc[t][r];
            if (RELU) v = fmaxf(v, 0.0f);
            out[(long long)m * DOUT + n] = v;
        }
    }
}

// ---------------------------------------------------------------------------
// Host launcher
// ---------------------------------------------------------------------------
extern "C" void kernel_launch(void* const* d_in, const int* in_sizes, int n_in,
                              void* d_out, int out_size, void* d_ws, size_t ws_size,
                              hipStream_t stream) {
    const float* feat  = (const float*)d_in[0];   // [100000,128]
    const float* Ws1   = (const float*)d_in[1];   // [128,128]
    const float* Wn1   = (const float*)d_in[2];   // [128,128]
    const float* Ws2   = (const float*)d_in[3];   // [128,64]
    const float* Wn2   = (const float*)d_in[4];   // [128,64]
    const int*   esrc  = (const int*)d_in[5];     // [800000]
    const int*   edst  = (const int*)d_in[6];     // [800000]
    float*       out   = (float*)d_out;           // [100000,64]

    // Workspace layout (fp32): neigh_sum | deg | h1  => ~102.8 MB
    float* ns  = (float*)d_ws;                                  // 12.8M
    float* deg = ns + (size_t)N_NODES * D_FEAT;                 // 100K
    float* h1  = deg + N_NODES;                                 // 12.8M

    const int ZB = 256;
    const int nsN4  = (N_NODES * D_FEAT) / 4;   // 3.2M float4
    const int degN4 = N_NODES / 4;              // 25K float4 (100000 % 4 == 0)

    // ---- Layer 1 ----
    zero4_kernel<<<(nsN4 + ZB - 1) / ZB, ZB, 0, stream>>>((float4*)ns, nsN4);
    zero4_kernel<<<(degN4 + ZB - 1) / ZB, ZB, 0, stream>>>((float4*)deg, degN4);

    {   // one wave per edge: 800000*32 threads / 256 = 100000 blocks
        dim3 grid((N_EDGES * 32) / 256);
        aggregate_kernel<true><<<grid, 256, 0, stream>>>(feat, esrc, edst, ns, deg, N_EDGES);
    }
    // M=32 per block: 100000/32 = 3125 exact -> zero divergence (EXEC all-1s)
    sage_layer_kernel<128, 2, true><<<N_NODES / 32, 256, 0, stream>>>(
        feat, ns, deg, Ws1, Wn1, h1);

    // ---- Layer 2 (deg unchanged: same edge_dst) ----
    zero4_kernel<<<(nsN4 + ZB - 1) / ZB, ZB, 0, stream>>>((float4*)ns, nsN4);
    {
        dim3 grid((N_EDGES * 32) / 256);
        aggregate_kernel<false><<<grid, 256, 0, stream>>>(h1, esrc, edst, ns, deg, N_EDGES);
    }
    sage_layer_kernel<64, 2, false><<<N_NODES / 32, 128, 0, stream>>>(
        h1, ns, deg, Ws2, Wn2, out);
}